// SparseAE_26551487824728
// MI455X (gfx1250) — compile-verified
//
#include <hip/hip_runtime.h>

typedef __attribute__((ext_vector_type(2))) float v2f;
typedef __attribute__((ext_vector_type(8))) float v8f;

#define NB 8192      // batch rows
#define ND 1024      // model dim (K of encoder GEMM)
#define NL 16384     // latent dim (N of encoder GEMM)
#define K_TOP 32
#define EPSV 1e-7f

// d_out layout (flat f32, return order): z, z_pre, x_rec, mu, std
#define Z_OFF    ((size_t)0)
#define ZPRE_OFF ((size_t)NB * NL)
#define XREC_OFF ((size_t)2 * NB * NL)
#define MU_OFF   (XREC_OFF + (size_t)NB * ND)
#define STD_OFF  (MU_OFF + (size_t)NB)

// ---------------------------------------------------------------------------
// CDNA5 async global->LDS copy (ASYNCcnt path, CDNA5 ISA).
// LDS destination comes from the VDST vgpr (low 32 bits of the LDS pointer).
// ---------------------------------------------------------------------------
__device__ __forceinline__ void async_ld_b128(unsigned lds_off,
                                              const float* gaddr) {
  asm volatile("global_load_async_to_lds_b128 %0, %1, off"
               :
               : "v"(lds_off), "v"(gaddr)
               : "memory");
}
#define WAIT_ASYNC_8() asm volatile("s_wait_asynccnt 0x8" ::: "memory")
#define WAIT_ASYNC_0() asm volatile("s_wait_asynccnt 0x0" ::: "memory")

// ---------------------------------------------------------------------------
// Kernel 1: per-row mean / std + fused xnb = (x-mu)/(std+eps) - tied_bias.
// One block (256 threads) per row; xnb goes to workspace so the GEMM's
// tile staging becomes a pure copy (async-DMA-able).
// ---------------------------------------------------------------------------
__global__ __launch_bounds__(256) void sae_norm_xnb(
    const float* __restrict__ x, const float* __restrict__ tb,
    float* __restrict__ mu_out, float* __restrict__ std_out,
    float* __restrict__ xnb) {
  __shared__ float s_sum[256];
  __shared__ float s_sq[256];
  __shared__ float s_mu, s_rs;
  const int row = blockIdx.x;
  const int tid = threadIdx.x;
  const float* xr = x + (size_t)row * ND;
  float v[4];
  float s = 0.f, q = 0.f;
#pragma unroll
  for (int i = 0; i < 4; ++i) {
    v[i] = xr[tid + i * 256];
    s += v[i];
    q += v[i] * v[i];
  }
  s_sum[tid] = s;
  s_sq[tid] = q;
  __syncthreads();
  for (int off = 128; off > 0; off >>= 1) {
    if (tid < off) {
      s_sum[tid] += s_sum[tid + off];
      s_sq[tid] += s_sq[tid + off];
    }
    __syncthreads();
  }
  if (tid == 0) {
    float mu = s_sum[0] * (1.0f / ND);
    float var = s_sq[0] * (1.0f / ND) - mu * mu;
    var = var < 0.f ? 0.f : var;
    float sd = sqrtf(var);
    mu_out[row] = mu;
    std_out[row] = sd;
    s_mu = mu;
    s_rs = 1.0f / (sd + EPSV);
  }
  __syncthreads();
  const float m = s_mu, r = s_rs;
  float* o = xnb + (size_t)row * ND;
#pragma unroll
  for (int i = 0; i < 4; ++i)
    o[tid + i * 256] = (v[i] - m) * r - tb[tid + i * 256];
}

// ---------------------------------------------------------------------------
// Kernel 2: encoder GEMM  z_pre = xnb @ W_enc^T   (f32 WMMA 16x16x4)
// Block tile 128(M) x 128(N) x 32(K), 8 waves, 64x32 per wave (4x2 tiles).
// Double-buffered LDS filled by global_load_async_to_lds_b128.
// ---------------------------------------------------------------------------
#define BM 128
#define BN 128
#define BK 32
#define LDT 36   // padded LDS row stride: 16B-aligned segments, conflict-free

__global__ __launch_bounds__(256) void sae_encoder_wmma(
    const float* __restrict__ xnb, const float* __restrict__ W_enc,
    float* __restrict__ z_pre) {
  __shared__ float As[2][BM * LDT];
  __shared__ float Bs[2][BN * LDT];

  const int tid = threadIdx.x;
  const int bn = blockIdx.x;   // N block (0..127)
  const int bm = blockIdx.y;   // M block (0..63)

  const int wave = tid >> 5;
  const int lane = tid & 31;
  const int lrow = lane & 15;
  const int lhi = lane >> 4;
  const int wm = (wave >> 2) * 64;    // wave M origin: 0 or 64
  const int wn = (wave & 3) * 32;     // wave N origin: 0,32,64,96

  v8f acc[4][2];
  {
    v8f vzero = {};
#pragma unroll
    for (int mt = 0; mt < 4; ++mt)
#pragma unroll
      for (int nt = 0; nt < 2; ++nt) acc[mt][nt] = vzero;
  }

  // staging geometry: 8 threads per row, 16B each -> 32 rows per pass
  const int trow = tid >> 3;          // 0..31
  const int tcol = (tid & 7) * 4;     // float index: 0,4,...,28

  const float* gA = xnb + (size_t)(bm * BM + trow) * ND + tcol;
  const float* gB = W_enc + (size_t)(bn * BN + trow) * ND + tcol;

  // issue chunk 0 into buffer 0 (8 async b128 per thread)
#pragma unroll
  for (int p = 0; p < 4; ++p) {
    async_ld_b128((unsigned)(uintptr_t)&As[0][(p * 32 + trow) * LDT + tcol],
                  gA + (size_t)p * 32 * ND);
    async_ld_b128((unsigned)(uintptr_t)&Bs[0][(p * 32 + trow) * LDT + tcol],
                  gB + (size_t)p * 32 * ND);
  }

  const int NC = ND / BK;  // 32 K-chunks
  for (int c = 0; c < NC; ++c) {
    const int cur = c & 1;
    if (c + 1 < NC) {
      const int nxt = cur ^ 1;
      const int ko = (c + 1) * BK;
#pragma unroll
      for (int p = 0; p < 4; ++p) {
        async_ld_b128(
            (unsigned)(uintptr_t)&As[nxt][(p * 32 + trow) * LDT + tcol],
            gA + (size_t)p * 32 * ND + ko);
        async_ld_b128(
            (unsigned)(uintptr_t)&Bs[nxt][(p * 32 + trow) * LDT + tcol],
            gB + (size_t)p * 32 * ND + ko);
      }
      WAIT_ASYNC_8();  // chunk c (in-order) complete; chunk c+1 in flight
    } else {
      WAIT_ASYNC_0();
    }
    __syncthreads();  // all waves' chunk-c tiles visible in LDS

    const float* Ab = &As[cur][0];
    const float* Bb = &Bs[cur][0];
#pragma unroll
    for (int kk = 0; kk < BK; kk += 4) {
      const int kb = kk + lhi * 2;   // even -> 8B-aligned ds_load_b64
      v2f a[4], b[2];
#pragma unroll
      for (int mt = 0; mt < 4; ++mt)
        a[mt] = *(const v2f*)&Ab[(wm + mt * 16 + lrow) * LDT + kb];
#pragma unroll
      for (int nt = 0; nt < 2; ++nt)
        b[nt] = *(const v2f*)&Bb[(wn + nt * 16 + lrow) * LDT + kb];
#pragma unroll
      for (int mt = 0; mt < 4; ++mt)
#pragma unroll
        for (int nt = 0; nt < 2; ++nt)
          acc[mt][nt] = __builtin_amdgcn_wmma_f32_16x16x4_f32(
              false, a[mt], false, b[nt], (short)0, acc[mt][nt], false, false);
    }
    __syncthreads();  // done reading buf[cur] before it is refilled
  }

  // Store: VGPR i, lanes<16 -> M=i, lanes>=16 -> M=i+8; N=lrow.
#pragma unroll
  for (int mt = 0; mt < 4; ++mt) {
#pragma unroll
    for (int nt = 0; nt < 2; ++nt) {
      int gm = bm * BM + wm + mt * 16 + lhi * 8;
      int gn = bn * BN + wn + nt * 16 + lrow;
      float* out = z_pre + (size_t)gm * NL + gn;
#pragma unroll
      for (int i = 0; i < 8; ++i) out[(size_t)i * NL] = acc[mt][nt][i];
    }
  }
}

// ---------------------------------------------------------------------------
// Kernel 3: exact k-th-largest via MSB-first radix select, row in registers.
// Writes z and compacts (idx,val) pairs for the sparse decoder.
// ---------------------------------------------------------------------------
__global__ __launch_bounds__(512) void sae_topk(
    const float* __restrict__ z_pre, float* __restrict__ z,
    int* __restrict__ wi, float* __restrict__ wv, int* __restrict__ wc) {
  __shared__ unsigned s_hist[256];
  __shared__ unsigned s_prefix;
  __shared__ int s_kneed;
  __shared__ int s_count;

  const int row = blockIdx.x;
  const int tid = threadIdx.x;
  const float* zr = z_pre + (size_t)row * NL;

  float v[32];
  unsigned key[32];
#pragma unroll
  for (int c = 0; c < 8; ++c) {
    float4 t = ((const float4*)zr)[c * 512 + tid];
    v[c * 4 + 0] = t.x;
    v[c * 4 + 1] = t.y;
    v[c * 4 + 2] = t.z;
    v[c * 4 + 3] = t.w;
  }
#pragma unroll
  for (int j = 0; j < 32; ++j) {
    unsigned u = __float_as_uint(v[j]);
    key[j] = (u & 0x80000000u) ? ~u : (u | 0x80000000u);
  }
  if (tid == 0) {
    s_prefix = 0u;
    s_kneed = K_TOP;
    s_count = 0;
  }
  for (int round = 0; round < 4; ++round) {
    const int shift = 24 - 8 * round;
    __syncthreads();
    if (tid < 256) s_hist[tid] = 0u;
    __syncthreads();
    const unsigned prefix = s_prefix;
    const unsigned pmask = round ? (0xFFFFFFFFu << (shift + 8)) : 0u;
#pragma unroll
    for (int j = 0; j < 32; ++j)
      if ((key[j] & pmask) == prefix)
        atomicAdd(&s_hist[(key[j] >> shift) & 255u], 1u);
    __syncthreads();
    if (tid == 0) {
      int kneed = s_kneed;
      unsigned cum = 0;
      int sel = 0;
      for (int b = 255; b >= 0; --b) {
        unsigned c = s_hist[b];
        if (cum + c >= (unsigned)kneed) {
          sel = b;
          break;
        }
        cum += c;
      }
      s_kneed = kneed - (int)cum;
      s_prefix = prefix | ((unsigned)sel << shift);
    }
  }
  __syncthreads();
  const unsigned tkey = s_prefix;
  const unsigned tu = (tkey & 0x80000000u) ? (tkey ^ 0x80000000u) : ~tkey;
  const float thresh = __uint_as_float(tu);

  float* zo = z + (size_t)row * NL;
#pragma unroll
  for (int c = 0; c < 8; ++c) {
    const int base = (c * 512 + tid) * 4;
    float4 o;
    o.x = (v[c * 4 + 0] >= thresh) ? v[c * 4 + 0] : 0.0f;
    o.y = (v[c * 4 + 1] >= thresh) ? v[c * 4 + 1] : 0.0f;
    o.z = (v[c * 4 + 2] >= thresh) ? v[c * 4 + 2] : 0.0f;
    o.w = (v[c * 4 + 3] >= thresh) ? v[c * 4 + 3] : 0.0f;
    ((float4*)zo)[c * 512 + tid] = o;
#pragma unroll
    for (int q = 0; q < 4; ++q) {
      float val = v[c * 4 + q];
      if (val >= thresh) {
        int pos = atomicAdd(&s_count, 1);
        if (pos < 64) {
          wi[row * 64 + pos] = base + q;
          wv[row * 64 + pos] = val;
        }
      }
    }
  }
  __syncthreads();
  if (tid == 0) wc[row] = s_count < 64 ? s_count : 64;
}

// ---------------------------------------------------------------------------
// Kernel 4: sparse decode  x_rec = (z @ W_dec^T + tb) * (std+eps) + mu
// Only ~32 active latents per row touch W_dec (L2-resident, 64 MB).
// ---------------------------------------------------------------------------
__global__ __launch_bounds__(256) void sae_decoder(
    const float* __restrict__ W_dec, const float* __restrict__ tb,
    const float* __restrict__ mu, const float* __restrict__ stdv,
    const int* __restrict__ wi, const float* __restrict__ wv,
    const int* __restrict__ wc, float* __restrict__ x_rec) {
  __shared__ int s_idx[64];
  __shared__ float s_val[64];
  const int row = blockIdx.x;
  const int tid = threadIdx.x;
  int cnt = wc[row];
  cnt = cnt < 64 ? cnt : 64;
  if (tid < 64) {
    s_idx[tid] = wi[row * 64 + tid];
    s_val[tid] = wv[row * 64 + tid];
  }
  __syncthreads();
  const float m = mu[row];
  const float sd = stdv[row] + EPSV;
  float* out = x_rec + (size_t)row * ND;
#pragma unroll
  for (int i = 0; i < 4; ++i) {
    const int d = tid + i * 256;
    float acc = tb[d];
    for (int j = 0; j < cnt; ++j)
      acc = fmaf(s_val[j], W_dec[(size_t)d * NL + s_idx[j]], acc);
    out[d] = acc * sd + m;
  }
}

// ---------------------------------------------------------------------------
extern "C" void kernel_launch(void* const* d_in, const int* in_sizes, int n_in,
                              void* d_out, int out_size, void* d_ws,
                              size_t ws_size, hipStream_t stream) {
  const float* x = (const float*)d_in[0];
  const float* W_enc = (const float*)d_in[1];
  const float* W_dec = (const float*)d_in[2];
  const float* tb = (const float*)d_in[3];

  float* out = (float*)d_out;
  float* z = out + Z_OFF;
  float* z_pre = out + ZPRE_OFF;
  float* x_rec = out + XREC_OFF;
  float* mu = out + MU_OFF;
  float* stdv = out + STD_OFF;

  // workspace: xnb (32MB) + compacted sparse pairs per row + counts
  const size_t XNB_BYTES = (size_t)NB * ND * 4;
  float* ws_xnb = (float*)d_ws;
  int* ws_idx = (int*)((char*)d_ws + XNB_BYTES);
  float* ws_val = (float*)((char*)d_ws + XNB_BYTES + (size_t)NB * 64 * 4);
  int* ws_cnt = (int*)((char*)d_ws + XNB_BYTES + (size_t)2 * NB * 64 * 4);

  sae_norm_xnb<<<NB, 256, 0, stream>>>(x, tb, mu, stdv, ws_xnb);

  dim3 grid(NL / BN, NB / BM);
  sae_encoder_wmma<<<grid, 256, 0, stream>>>(ws_xnb, W_enc, z_pre);

  sae_topk<<<NB, 512, 0, stream>>>(z_pre, z, ws_idx, ws_val, ws_cnt);

  sae_decoder<<<NB, 256, 0, stream>>>(W_dec, tb, mu, stdv, ws_idx, ws_val,
                                      ws_cnt, x_rec);
}